// SeparableFiberBundleConv_36541581754380
// MI455X (gfx1250) — compile-verified
//
#include <hip/hip_runtime.h>

typedef __attribute__((ext_vector_type(2))) float v2f;
typedef __attribute__((ext_vector_type(8))) float v8f;

#define N_NODES 10000
#define O_DIM   8
#define C_DIM   128
#define E_DIM   160000
#define B_DIM   64
#define OC      (O_DIM * C_DIM)   // 1024

// ---------------- zero x_1 workspace (10.24M floats = 2.56M float4) ------------
__global__ void zero_ws_kernel(float4* __restrict__ p) {
    int i = blockIdx.x * blockDim.x + threadIdx.x;
    p[i] = make_float4(0.f, 0.f, 0.f, 0.f);
}

// ---------------- fiber_kernel[p,o,c] = sum_b fkb[p,o,b] * Wf[b,c] -------------
__global__ void fiber_precompute_kernel(const float* __restrict__ fkb,
                                        const float* __restrict__ Wf,
                                        float* __restrict__ fk) {
    int idx = blockIdx.x * blockDim.x + threadIdx.x;   // 0 .. 8*8*128-1
    int c  = idx & (C_DIM - 1);
    int po = idx >> 7;                                  // p*8 + o
    const float* row = fkb + po * B_DIM;
    float s = 0.f;
#pragma unroll
    for (int b = 0; b < B_DIM; ++b) s += row[b] * Wf[b * C_DIM + c];
    fk[idx] = s;
}

// ---------------- fused edge kernel: WMMA GEMM + gather*mul + scatter-add ------
// rows r = e*8 + o over [E*O, B] @ W_kernel[B, C]; one wave = 16 rows x 128 cols.
__global__ __launch_bounds__(256, 2)
void edge_wmma_kernel(const float* __restrict__ x,
                      const float* __restrict__ kb,
                      const int*   __restrict__ ei,
                      const float* __restrict__ Wk,
                      float*       __restrict__ x1) {
    __shared__ float Wlds[B_DIM * C_DIM];          // 32 KB
    for (int i = threadIdx.x; i < B_DIM * C_DIM; i += 256) Wlds[i] = Wk[i];
    __syncthreads();

    const int lane  = threadIdx.x & 31;
    const int wave  = threadIdx.x >> 5;
    const int m     = lane & 15;                   // row/col within 16
    const int khalf = lane >> 4;                   // K-half select per ISA layout

    const int tile     = blockIdx.x * 8 + wave;    // 80000 tiles total
    const int base_row = tile * 16;

    // Preload A fragments: 16 K-steps of 4, v2f each (ISA 16x4 f32 layout:
    // VGPR0 = K {0|2}, VGPR1 = K {1|3} per lane half).
    v2f afrag[16];
    const float* arow = kb + (size_t)(base_row + m) * B_DIM + 2 * khalf;
#pragma unroll
    for (int ks = 0; ks < 16; ++ks) {
        v2f a;
        a.x = arow[ks * 4 + 0];
        a.y = arow[ks * 4 + 1];
        afrag[ks] = a;
    }

    // Per-lane output rows: M = i + 8*khalf (C/D layout). Precompute offsets.
    int xoff[8], doff[8];
#pragma unroll
    for (int i = 0; i < 8; ++i) {
        int g   = base_row + i + 8 * khalf;        // global (e,o) row
        int e   = g >> 3;
        int o   = g & 7;
        int src = ei[e];
        int dst = ei[E_DIM + e];
        xoff[i] = src * OC + o * C_DIM;
        doff[i] = dst * OC + o * C_DIM;
    }

#pragma unroll 1
    for (int t = 0; t < 8; ++t) {                  // 8 N-tiles of 16 columns
        const int col = t * 16 + m;

        // Issue the 8 x[src] gathers up front so their latency hides under
        // the 16 WMMAs of this tile (they do not depend on the accumulator).
        float xv[8];
#pragma unroll
        for (int i = 0; i < 8; ++i) xv[i] = x[xoff[i] + col];

        v8f acc = {};
#pragma unroll
        for (int ks = 0; ks < 16; ++ks) {
            const int k0 = ks * 4 + 2 * khalf;     // B layout mirrors A (row-striped)
            v2f b;
            b.x = Wlds[(k0 + 0) * C_DIM + col];
            b.y = Wlds[(k0 + 1) * C_DIM + col];
            acc = __builtin_amdgcn_wmma_f32_16x16x4_f32(
                false, afrag[ks], false, b, (short)0, acc, false, false);
        }

        // message = k * x[src]; scatter-add into x_1[dst]
#pragma unroll
        for (int i = 0; i < 8; ++i) {
            atomicAdd(&x1[doff[i] + col], acc[i] * xv[i]);
        }
    }
}

// ---------------- x_2[n,p,c] = (1/O) * sum_o x_1[n,o,c] * fk[p,o,c] ------------
__global__ void combine_kernel(const float* __restrict__ x1,
                               const float* __restrict__ fk,
                               float* __restrict__ out) {
    int tid = blockIdx.x * blockDim.x + threadIdx.x;  // over N*C
    int n = tid >> 7;
    int c = tid & (C_DIM - 1);
    float xv[8];
#pragma unroll
    for (int o = 0; o < 8; ++o) xv[o] = x1[n * OC + o * C_DIM + c];
#pragma unroll
    for (int p = 0; p < 8; ++p) {
        float s = 0.f;
#pragma unroll
        for (int o = 0; o < 8; ++o) s += xv[o] * fk[p * OC + o * C_DIM + c];
        out[n * OC + p * C_DIM + c] = s * 0.125f;
    }
}

extern "C" void kernel_launch(void* const* d_in, const int* in_sizes, int n_in,
                              void* d_out, int out_size, void* d_ws, size_t ws_size,
                              hipStream_t stream) {
    const float* x   = (const float*)d_in[0];   // [N, O, C]
    const float* kb  = (const float*)d_in[1];   // [E, O, B]
    const float* fkb = (const float*)d_in[2];   // [O, O, B]
    const int*   ei  = (const int*)  d_in[3];   // [2, E]
    const float* Wk  = (const float*)d_in[4];   // [B, C]
    const float* Wf  = (const float*)d_in[5];   // [B, C]
    // d_in[6] = bias, intentionally unused (reference does not add it)
    float* out = (float*)d_out;

    float* x1 = (float*)d_ws;                           // N*O*C floats (40.96 MB)
    float* fk = x1 + (size_t)N_NODES * OC;              // O*O*C floats (32 KB)

    // 1) zero x_1 accumulator
    zero_ws_kernel<<<(N_NODES * OC / 4) / 256, 256, 0, stream>>>((float4*)x1);
    // 2) fiber kernel precompute (8*8*128 = 8192 elems)
    fiber_precompute_kernel<<<(O_DIM * OC) / 256, 256, 0, stream>>>(fkb, Wf, fk);
    // 3) fused GEMM + gather*mul + scatter-add (E*O/16 = 80000 wave-tiles, 8/block)
    edge_wmma_kernel<<<(E_DIM * O_DIM / 16) / 8, 256, 0, stream>>>(x, kb, ei, Wk, x1);
    // 4) orientation mixing (N*C threads)
    combine_kernel<<<(N_NODES * C_DIM) / 256, 256, 0, stream>>>(x1, fk, out);
}